// HashSAT_27745488733058
// MI455X (gfx1250) — compile-verified
//
#include <hip/hip_runtime.h>
#include <hip/hip_bf16.h>
#include <math.h>

// Problem constants (match reference)
#define NNODES 100000
#define NEDGES 3200000
#define HID    512
#define OUTC   32

typedef float v2f __attribute__((ext_vector_type(2)));
typedef float v8f __attribute__((ext_vector_type(8)));

#define LDA 516          // padded LDS row stride (floats): 516 % 64 = 4 -> conflict-free frag reads
#define GEMM_THREADS 256 // 8 waves
#define NW 8             // waves per gemm block

// ---------------------------------------------------------------------------
// zero fill (float4 granularity; all our sizes are multiples of 4)
// ---------------------------------------------------------------------------
__global__ __launch_bounds__(256) void zero_kernel(float* __restrict__ p, long n4) {
    long i = (long)blockIdx.x * blockDim.x + threadIdx.x;
    if (i < n4) {
        float4 z = make_float4(0.f, 0.f, 0.f, 0.f);
        ((float4*)p)[i] = z;
    }
}

// ---------------------------------------------------------------------------
// degree accumulation: deg_out from src, deg_in from dst
// ---------------------------------------------------------------------------
__global__ __launch_bounds__(256) void degree_kernel(const int* __restrict__ src,
                                                     const int* __restrict__ dst,
                                                     float* __restrict__ degOut,
                                                     float* __restrict__ degIn) {
    int e = blockIdx.x * blockDim.x + threadIdx.x;
    if (e < NEDGES) {
        atomicAdd(&degOut[src[e]], 1.0f);
        atomicAdd(&degIn[dst[e]], 1.0f);
    }
}

// in-place: deg -> rsqrt(max(deg,1))
__global__ __launch_bounds__(256) void invsqrt_kernel(float* __restrict__ degOut,
                                                      float* __restrict__ degIn) {
    int i = blockIdx.x * blockDim.x + threadIdx.x;
    if (i < NNODES) {
        degOut[i] = rsqrtf(fmaxf(degOut[i], 1.0f));
        degIn[i]  = rsqrtf(fmaxf(degIn[i], 1.0f));
    }
}

// ---------------------------------------------------------------------------
// sparse aggregation: agg[dst[e]] += H[src[e]] * invOut[src[e]]
// one wave (32 lanes) per edge; 512 floats/row via float4 streams,
// contiguous-dword atomic adds (L2-resident).
// ---------------------------------------------------------------------------
__global__ __launch_bounds__(256) void aggregate_kernel(const int* __restrict__ src,
                                                        const int* __restrict__ dst,
                                                        const float* __restrict__ H,
                                                        const float* __restrict__ invOut,
                                                        float* __restrict__ agg) {
    int gwave = (int)((blockIdx.x * blockDim.x + threadIdx.x) >> 5);
    int lane  = threadIdx.x & 31;
    if (gwave >= NEDGES) return;
    int s = src[gwave];
    int d = dst[gwave];
    float sc = invOut[s];
    const float4* hs = (const float4*)(H + (long)s * HID) + lane;
    float* ad = agg + (long)d * HID + lane * 4;
    #pragma unroll
    for (int it = 0; it < HID / 128; ++it) {
        float4 v = hs[it * 32];
        float* p = ad + it * 128;
        atomicAdd(p + 0, v.x * sc);
        atomicAdd(p + 1, v.y * sc);
        atomicAdd(p + 2, v.z * sc);
        atomicAdd(p + 3, v.w * sc);
    }
}

// ---------------------------------------------------------------------------
// GEMM: Out[r, c] = act( sum_k (A[r,k] * invIn[r]) * W[k,c] + bias[c] )
//   A: [NNODES, HID] (lda = HID), W: [HID, NCOLS], Out: [NNODES, NCOLS]
// One block = 16-row panel staged (scaled) into LDS. 8 waves cover the
// 16-column tiles; each wave drives TWO independent v_wmma_f32_16x16x4_f32
// accumulation chains (tiles t and t+NW) sharing one A fragment per K-step.
// NCOLS is a template constant so all B loads / C stores use immediate
// offsets from a per-lane base pointer (no per-load address VALU).
// In-place safe (Out may alias A): all global A reads precede the barrier,
// each block writes only its own rows. ACT: 0 = relu, 1 = sigmoid.
// ---------------------------------------------------------------------------
template <int NCOLS, int ACT>
__global__ __launch_bounds__(GEMM_THREADS) void gemm_kernel(const float* A,
                                                            const float* __restrict__ invIn,
                                                            const float* __restrict__ W,
                                                            const float* __restrict__ bias,
                                                            float* Out) {
    __shared__ float As[16 * LDA];

    const int r0 = blockIdx.x * 16;

    // --- stage A panel (16 x 512) into LDS, fused invIn row scale ---
    for (int i = threadIdx.x * 4; i < 16 * HID; i += GEMM_THREADS * 4) {
        int row  = i >> 9;        // / 512
        int colc = i & (HID - 1); // % 512
        float sc = invIn[r0 + row];
        float4 v = *(const float4*)(A + (long)(r0 + row) * HID + colc);
        float* p = &As[row * LDA + colc];
        p[0] = v.x * sc;
        p[1] = v.y * sc;
        p[2] = v.z * sc;
        p[3] = v.w * sc;
    }
    __syncthreads();

    const int wave = threadIdx.x >> 5;
    const int lane = threadIdx.x & 31;
    const int m    = lane & 15;  // A's M index == B's N index
    const int hi   = lane >> 4;  // K sub-select (0 -> K 0,1 ; 1 -> K 2,3)

    const float* Arow = &As[m * LDA + 2 * hi];
    constexpr int ntiles = NCOLS / 16;

    for (int t0 = wave; t0 < ntiles; t0 += 2 * NW) {
        const int col0 = t0 * 16 + m;

        if (t0 + NW < ntiles) {
            // ---- dual-tile path: two independent WMMA chains ----
            const int col1 = (t0 + NW) * 16 + m;
            const float* W0 = W + col0 + 2 * hi * NCOLS;
            const float* W1 = W + col1 + 2 * hi * NCOLS;

            float bv0 = bias[col0], bv1 = bias[col1];
            v8f acc0, acc1;
            #pragma unroll
            for (int j = 0; j < 8; ++j) { acc0[j] = bv0; acc1[j] = bv1; }

            #pragma unroll 16
            for (int k = 0; k < HID; k += 4) {
                v2f a;
                a.x = Arow[k];
                a.y = Arow[k + 1];
                v2f b0, b1;
                b0.x = W0[k * NCOLS];
                b0.y = W0[(k + 1) * NCOLS];
                b1.x = W1[k * NCOLS];
                b1.y = W1[(k + 1) * NCOLS];
                acc0 = __builtin_amdgcn_wmma_f32_16x16x4_f32(false, a, false, b0,
                                                             (short)0, acc0, false, false);
                acc1 = __builtin_amdgcn_wmma_f32_16x16x4_f32(false, a, false, b1,
                                                             (short)0, acc1, false, false);
            }

            float* O0 = Out + (long)(r0 + 8 * hi) * NCOLS + col0;
            float* O1 = Out + (long)(r0 + 8 * hi) * NCOLS + col1;
            #pragma unroll
            for (int j = 0; j < 8; ++j) {
                float v0 = acc0[j], v1 = acc1[j];
                if (ACT == 0) {
                    v0 = fmaxf(v0, 0.0f);
                    v1 = fmaxf(v1, 0.0f);
                } else {
                    v0 = 1.0f / (1.0f + expf(-v0));
                    v1 = 1.0f / (1.0f + expf(-v1));
                }
                O0[j * NCOLS] = v0;
                O1[j * NCOLS] = v1;
            }
        } else {
            // ---- single-tile path (used for NCOLS=32) ----
            const float* W0 = W + col0 + 2 * hi * NCOLS;
            float bv0 = bias[col0];
            v8f acc0;
            #pragma unroll
            for (int j = 0; j < 8; ++j) acc0[j] = bv0;

            #pragma unroll 16
            for (int k = 0; k < HID; k += 4) {
                v2f a;
                a.x = Arow[k];
                a.y = Arow[k + 1];
                v2f b0;
                b0.x = W0[k * NCOLS];
                b0.y = W0[(k + 1) * NCOLS];
                acc0 = __builtin_amdgcn_wmma_f32_16x16x4_f32(false, a, false, b0,
                                                             (short)0, acc0, false, false);
            }

            float* O0 = Out + (long)(r0 + 8 * hi) * NCOLS + col0;
            #pragma unroll
            for (int j = 0; j < 8; ++j) {
                float v0 = acc0[j];
                if (ACT == 0) {
                    v0 = fmaxf(v0, 0.0f);
                } else {
                    v0 = 1.0f / (1.0f + expf(-v0));
                }
                O0[j * NCOLS] = v0;
            }
        }
    }
}

// ---------------------------------------------------------------------------
// host-side orchestration
// ---------------------------------------------------------------------------
static inline int cdiv(long a, long b) { return (int)((a + b - 1) / b); }

extern "C" void kernel_launch(void* const* d_in, const int* in_sizes, int n_in,
                              void* d_out, int out_size, void* d_ws, size_t ws_size,
                              hipStream_t stream) {
    const int*   src      = (const int*)d_in[0];
    const int*   dst      = (const int*)d_in[1];
    const float* H0       = (const float*)d_in[2];
    const float* W_hidden = (const float*)d_in[3]; // [3,512,512]
    const float* b_hidden = (const float*)d_in[4]; // [3,512]
    const float* W_out    = (const float*)d_in[5]; // [512,32]
    const float* b_out    = (const float*)d_in[6]; // [32]
    float*       out      = (float*)d_out;         // [N,32]

    // workspace layout
    float* invOut = (float*)d_ws;                 // N (deg_out -> rsqrt)
    float* invIn  = invOut + NNODES;              // N (deg_in  -> rsqrt)
    float* bufA   = invIn + NNODES;               // N*512
    float* bufB   = bufA + (long)NNODES * HID;    // N*512

    const long NH = (long)NNODES * HID;

    // 1) degrees -> inv sqrt
    zero_kernel<<<cdiv(2L * NNODES / 4, 256), 256, 0, stream>>>(invOut, 2L * NNODES / 4);
    degree_kernel<<<cdiv(NEDGES, 256), 256, 0, stream>>>(src, dst, invOut, invIn);
    invsqrt_kernel<<<cdiv(NNODES, 256), 256, 0, stream>>>(invOut, invIn);

    const int aggGrid  = cdiv((long)NEDGES * 32, 256); // one wave per edge
    const int gemmGrid = NNODES / 16;                  // 6250, exact

    // 2) layer 0: H0 -> bufA (agg), gemm in place -> H1 in bufA
    zero_kernel<<<cdiv(NH / 4, 256), 256, 0, stream>>>(bufA, NH / 4);
    aggregate_kernel<<<aggGrid, 256, 0, stream>>>(src, dst, H0, invOut, bufA);
    gemm_kernel<HID, 0><<<gemmGrid, GEMM_THREADS, 0, stream>>>(
        bufA, invIn, W_hidden + 0L * HID * HID, b_hidden + 0 * HID, bufA);

    // 3) layer 1: bufA -> bufB, in place -> H2 in bufB
    zero_kernel<<<cdiv(NH / 4, 256), 256, 0, stream>>>(bufB, NH / 4);
    aggregate_kernel<<<aggGrid, 256, 0, stream>>>(src, dst, bufA, invOut, bufB);
    gemm_kernel<HID, 0><<<gemmGrid, GEMM_THREADS, 0, stream>>>(
        bufB, invIn, W_hidden + 1L * HID * HID, b_hidden + 1 * HID, bufB);

    // 4) layer 2: bufB -> bufA, in place -> H3 in bufA
    zero_kernel<<<cdiv(NH / 4, 256), 256, 0, stream>>>(bufA, NH / 4);
    aggregate_kernel<<<aggGrid, 256, 0, stream>>>(src, dst, bufB, invOut, bufA);
    gemm_kernel<HID, 0><<<gemmGrid, GEMM_THREADS, 0, stream>>>(
        bufA, invIn, W_hidden + 2L * HID * HID, b_hidden + 2 * HID, bufA);

    // 5) output layer: bufA -> bufB (agg), gemm -> d_out [N,32], sigmoid
    zero_kernel<<<cdiv(NH / 4, 256), 256, 0, stream>>>(bufB, NH / 4);
    aggregate_kernel<<<aggGrid, 256, 0, stream>>>(src, dst, bufA, invOut, bufB);
    gemm_kernel<OUTC, 1><<<gemmGrid, GEMM_THREADS, 0, stream>>>(
        bufB, invIn, W_out, b_out, out);
}